// SineGen_48868137894412
// MI455X (gfx1250) — compile-verified
//
#include <hip/hip_runtime.h>

typedef float v2f __attribute__((ext_vector_type(2)));
typedef float v4f __attribute__((ext_vector_type(4)));
typedef float v8f __attribute__((ext_vector_type(8)));

#define BB   8
#define LL   2000
#define UPP  400
#define DIM  9
#define TT   (LL * UPP)                 // 800000
#define N_SINE  (BB * TT * DIM)         // 57,600,000
#define N_UV    (BB * TT)               // 6,400,000
#define OFF_UV  N_SINE
#define OFF_NZ  (N_SINE + N_UV)         // 64,000,000

__device__ __forceinline__ float fracf_(float x) { return x - floorf(x); }

// ---------------------------------------------------------------------------
// Kernel 1: per-(b,d) mod-1 exclusive prefix of r'_l = frac(400*rad_l) using
// V_WMMA_F32_16X16X4_F32 matmul-scan. One wave32 per (b,d); 8 chunks of 256.
// Chunk laid out as 16x16 (time down columns): X[m][n] = r'[l0 + m + 16n].
//   colScan = L(16x16, lower-tri ones) * X  = sum_c G_c(16x4) * X_rows(4x16)
//   offsets = sum_g A_g(colsum bcast)  * U_g(strict-upper indicator)
// Writes P[l] = frac(exclusive prefix), R[l] = rad_l (incl. rand_ini @ l==0).
// ---------------------------------------------------------------------------
__global__ __launch_bounds__(32) void sine_scan_wmma_kernel(
    const float* __restrict__ f0, const float* __restrict__ rand_ini,
    float* __restrict__ Ptab, float* __restrict__ Rtab)
{
  const int bd   = blockIdx.x;            // 0..71
  const int b    = bd / DIM, d = bd % DIM;
  const int lane = threadIdx.x;           // 0..31
  const bool lo  = lane < 16;
  const int  nl  = lo ? lane : lane - 16; // N (column) index / M index

  const float harm = (float)(d + 1);
  const float ri   = (d == 0) ? 0.0f : rand_ini[b * DIM + d];
  const float* f0b = f0 + b * LL;
  float* P = Ptab + bd * LL;
  float* R = Rtab + bd * LL;

  // Constant triangular A slices: G_c[m][k] = (m >= 4c+k).
  // A layout (16x4 f32): lane<16 -> {A[m][0],A[m][1]}, lane>=16 -> {A[m][2],A[m][3]}, m=nl.
  v2f G[4];
#pragma unroll
  for (int c = 0; c < 4; ++c) {
    G[c].x = (nl >= 4 * c + (lo ? 0 : 2)) ? 1.0f : 0.0f;
    G[c].y = (nl >= 4 * c + (lo ? 1 : 3)) ? 1.0f : 0.0f;
  }
  // Constant strict-upper indicator B: U_g[k][n] = ((4g+k) < n).
  // B layout (4x16 f32): lane<16 -> {B[0][n],B[1][n]}, lane>=16 -> {B[2][n],B[3][n]}, n=nl.
  v2f U[4];
#pragma unroll
  for (int g = 0; g < 4; ++g) {
    U[g].x = ((4 * g + (lo ? 0 : 2)) < nl) ? 1.0f : 0.0f;
    U[g].y = ((4 * g + (lo ? 1 : 3)) < nl) ? 1.0f : 0.0f;
  }

  if (lane == 0) P[0] = 0.0f;   // exclusive prefix at l=0

  float carry = 0.0f;
  for (int chunk = 0; chunk < 8; ++chunk) {
    const int l0 = chunk * 256;

    // Build data B-operand slices (rows 4c..4c+3 of X) and emit R/r'.
    v2f Bs[4];
#pragma unroll
    for (int c = 0; c < 4; ++c) {
      const int la = l0 + 4 * c + (lo ? 0 : 2) + 16 * nl;  // .x element
      float rv0 = 0.0f, rv1 = 0.0f;
      if (la < LL) {
        float rad = fracf_(f0b[la] * harm * (1.0f / 40000.0f));
        if (la == 0) rad += ri;
        R[la] = rad;
        double rp = (double)rad * 400.0;
        rv0 = (float)(rp - floor(rp));
      }
      if (la + 1 < LL) {
        float rad = fracf_(f0b[la + 1] * harm * (1.0f / 40000.0f));
        R[la + 1] = rad;
        double rp = (double)rad * 400.0;
        rv1 = (float)(rp - floor(rp));
      }
      Bs[c].x = rv0; Bs[c].y = rv1;
    }

    // Intra-column inclusive scan: acc = sum_c G_c * Bs_c
    v8f acc = {};
#pragma unroll
    for (int c = 0; c < 4; ++c)
      acc = __builtin_amdgcn_wmma_f32_16x16x4_f32(
          false, G[c], false, Bs[c], (short)0, acc, false, false);

    // Column sums live at row 15: acc[7], lanes 16..31.
    float cs[16];
#pragma unroll
    for (int c = 0; c < 16; ++c) cs[c] = __shfl(acc[7], 16 + c, 32);

    // Cross-column offsets: acc += sum_g bcast(colsum) * U_g
#pragma unroll
    for (int g = 0; g < 4; ++g) {
      v2f Ag;
      Ag.x = lo ? cs[4 * g + 0] : cs[4 * g + 2];
      Ag.y = lo ? cs[4 * g + 1] : cs[4 * g + 3];
      acc = __builtin_amdgcn_wmma_f32_16x16x4_f32(
          false, Ag, false, U[g], (short)0, acc, false, false);
    }

    // acc[m][n] + carry = inclusive prefix of element e = m+16n.
    // Exclusive prefix for element e+1 -> store frac at l = l0+e+1.
#pragma unroll
    for (int r = 0; r < 8; ++r) {
      const int m = lo ? r : r + 8;
      const int l = l0 + m + 16 * nl + 1;
      const float v = acc[r] + carry;
      if (l < LL) P[l] = fracf_(v);
    }
    carry = fracf_(__shfl(acc[7] + carry, 31, 32));  // element 255 inclusive
  }
}

// ---------------------------------------------------------------------------
// Kernel 2: streaming main kernel. 4 consecutive flat elements per thread.
// Streams (noise_in, sine, noise) use non-temporal b128 accesses so the
// reused 1.2 MB phase tables stay hot in WGP$/L2.
// ---------------------------------------------------------------------------
__global__ __launch_bounds__(256) void sine_main_kernel(
    const float* __restrict__ f0, const float* __restrict__ noise_in,
    const float* __restrict__ Ptab, const float* __restrict__ Rtab,
    float* __restrict__ out)
{
  const unsigned g = blockIdx.x * 256u + threadIdx.x;
  if (g >= (unsigned)(N_SINE / 4)) return;
  const unsigned i0 = g * 4u;

  const v4f nz4 = __builtin_nontemporal_load((const v4f*)noise_in + g);
  v4f sine4, noi4;

#pragma unroll
  for (int q = 0; q < 4; ++q) {
    const unsigned i  = i0 + (unsigned)q;
    const unsigned dd = i % (unsigned)DIM;
    const unsigned bt = i / (unsigned)DIM;
    const unsigned b  = bt / (unsigned)TT;
    const unsigned t  = bt % (unsigned)TT;
    const unsigned l  = t / (unsigned)UPP;
    const unsigned j  = t - l * (unsigned)UPP;

    const unsigned li = (b * (unsigned)DIM + dd) * (unsigned)LL + l;
    const float Pv  = Ptab[li];
    const float rv  = Rtab[li];
    const float f0v = f0[b * (unsigned)LL + l];

    float ph = fmaf((float)(j + 1u), rv, Pv);
    ph -= floorf(ph);
    const float s = 0.1f * __sinf(6.283185307179586f * ph);

    const bool  v    = f0v > 0.0f;
    const float namp = v ? 0.003f : (0.1f / 3.0f);
    const float nv   = namp * nz4[q];
    noi4[q]  = nv;
    sine4[q] = v ? (s + nv) : nv;
  }

  __builtin_nontemporal_store(sine4, (v4f*)out + g);
  __builtin_nontemporal_store(noi4, (v4f*)(out + OFF_NZ) + g);
}

// ---------------------------------------------------------------------------
// Kernel 3: uv output (B*T floats), non-temporal float4 stores.
// ---------------------------------------------------------------------------
__global__ __launch_bounds__(256) void sine_uv_kernel(
    const float* __restrict__ f0, float* __restrict__ out_uv)
{
  const unsigned g = blockIdx.x * 256u + threadIdx.x;
  if (g >= (unsigned)(N_UV / 4)) return;
  const unsigned i0 = g * 4u;
  v4f u;
#pragma unroll
  for (int q = 0; q < 4; ++q) {
    const unsigned i = i0 + (unsigned)q;
    const unsigned b = i / (unsigned)TT;
    const unsigned t = i % (unsigned)TT;
    const unsigned l = t / (unsigned)UPP;
    u[q] = (f0[b * (unsigned)LL + l] > 0.0f) ? 1.0f : 0.0f;
  }
  __builtin_nontemporal_store(u, (v4f*)out_uv + g);
}

extern "C" void kernel_launch(void* const* d_in, const int* in_sizes, int n_in,
                              void* d_out, int out_size, void* d_ws, size_t ws_size,
                              hipStream_t stream) {
  (void)in_sizes; (void)n_in; (void)out_size; (void)ws_size;
  const float* f0       = (const float*)d_in[0];
  const float* rand_ini = (const float*)d_in[1];
  const float* noise_in = (const float*)d_in[2];
  float* out = (float*)d_out;

  float* Ptab = (float*)d_ws;                 // 72*2000 floats
  float* Rtab = Ptab + BB * DIM * LL;         // 72*2000 floats (1.15 MB total)

  sine_scan_wmma_kernel<<<BB * DIM, 32, 0, stream>>>(f0, rand_ini, Ptab, Rtab);

  sine_uv_kernel<<<(N_UV / 4 + 255) / 256, 256, 0, stream>>>(f0, out + OFF_UV);

  sine_main_kernel<<<(N_SINE / 4 + 255) / 256, 256, 0, stream>>>(
      f0, noise_in, Ptab, Rtab, out);
}